// MoEGate_91122026152203
// MI455X (gfx1250) — compile-verified
//
#include <hip/hip_runtime.h>
#include <math.h>

// Problem constants (from reference): B=4, S=4096, H=2048, E=64, TOPK=8
#define TOKENS      16384
#define HDIM        2048
#define NEXP        64
#define NCOL        128      // 64 transform cols + 64 noise cols, fused GEMM
#define KC          32       // K-chunk staged in LDS by the TDM
#define NCHUNK      (HDIM / KC)              // 64
#define LDST        36       // KC + 4 pad dwords (TDM pad_enable), 16B aligned
#define STAGE_FLTS  (NCOL * LDST)            // 4608 floats per buffer
#define TPB         256      // 8 waves (wave32)
#define TOK_PER_BLK 128      // 8 waves * 16 tokens
#define NBLOCKS     (TOKENS / TOK_PER_BLK)   // 128
#define TILE_ROW    66       // padded per-token row stride in reused LDS tile buffer
#define TILE_WAVE   (16 * TILE_ROW)

typedef float v2f __attribute__((ext_vector_type(2)));
typedef float v8f __attribute__((ext_vector_type(8)));
typedef unsigned int u32x4 __attribute__((ext_vector_type(4)));
typedef int i32x4 __attribute__((ext_vector_type(4)));
typedef int i32x8 __attribute__((ext_vector_type(8)));

// Issue one TDM 2D tile load: 64 rows (experts) x KC dwords, global stride HDIM,
// into LDS at byte offset lds_off with 4-dword padding after every KC dwords.
__device__ __forceinline__ void tdm_load_rows(unsigned lds_off, const float* gsrc) {
    unsigned long long ga = (unsigned long long)gsrc;
    u32x4 g0;
    g0[0] = 1u;                                         // count=1, load, user mode
    g0[1] = lds_off;                                    // lds_addr (bytes)
    g0[2] = (unsigned)(ga & 0xFFFFFFFFull);             // global_addr[31:0]
    g0[3] = (unsigned)((ga >> 32) & 0x01FFFFFFull)      // global_addr[56:32]
          | (2u << 30);                                 // type = 2 ("image")
    i32x8 g1;
    g1[0] = (int)((2u << 16)      // data_size = 4 bytes
                | (1u << 20)      // pad_enable
                | (4u << 22)      // pad_interval: 2^(4+1) = 32 dwords
                | (3u << 25));    // pad_amount: 3+1 = 4 dwords  -> stride 36
    g1[1] = (int)((HDIM & 0xFFFF) << 16);                       // tensor_dim0 lo16
    g1[2] = (int)(((HDIM >> 16) & 0xFFFF) | ((NEXP & 0xFFFF) << 16)); // dim0 hi | dim1 lo
    g1[3] = (int)((KC & 0xFFFF) << 16);                         // dim1 hi=0 | tile_dim0=KC
    g1[4] = (int)(NEXP & 0xFFFF);                               // tile_dim1=64, tile_dim2=0
    g1[5] = (int)HDIM;                                          // tensor_dim0_stride lo32
    g1[6] = 0;                                                  // stride hi | dim1_stride lo
    g1[7] = 0;
    i32x4 z4 = {0, 0, 0, 0};
#if __clang_major__ >= 23
    i32x8 z8 = {0, 0, 0, 0, 0, 0, 0, 0};
    __builtin_amdgcn_tensor_load_to_lds(g0, g1, z4, z4, z8, 0);
#else
    __builtin_amdgcn_tensor_load_to_lds(g0, g1, z4, z4, 0);
#endif
}

__global__ void moe_zero_ws_91122026152203(float* __restrict__ ws) {
    if (threadIdx.x < 2 * NEXP) ws[threadIdx.x] = 0.0f;
}

__global__ void __launch_bounds__(TPB)
moe_gate_main_91122026152203(const float* __restrict__ x,
                             const float* __restrict__ wt,
                             const float* __restrict__ wn,
                             const float* __restrict__ noise,
                             float* __restrict__ gsum) {
    // Double-buffered TDM staging (2 x 128 cols x 36 floats); reused at the end
    // as the 8 x (16 token x 64 expert) combined-weight tiles (8448 floats).
    __shared__ float lds[2 * STAGE_FLTS];
    __shared__ float sh_sumw[NEXP];
    __shared__ float sh_soft[NEXP];

    const int tid  = threadIdx.x;
    const int wave = tid >> 5;
    const int lane = tid & 31;
    const int col  = lane & 15;   // N / M position within a 16-wide stripe
    const int khi  = lane >> 4;   // K-half select: lanes 16-31 hold K+2/K+3

    const int tokBase = blockIdx.x * TOK_PER_BLK + wave * 16;
    const float* xrow = x + (size_t)(tokBase + col) * HDIM + 2 * khi;

    const unsigned lds_base = (unsigned)(unsigned long long)(void*)&lds[0];

    const v8f vzero = {0.f, 0.f, 0.f, 0.f, 0.f, 0.f, 0.f, 0.f};
    v8f acc[8];
#pragma unroll
    for (int i = 0; i < 8; ++i) acc[i] = vzero;

    // Prime the pipeline: chunk 0 into buffer 0 (wave 0 drives the TDM).
    if (wave == 0) {
        tdm_load_rows(lds_base,                      wt);
        tdm_load_rows(lds_base + NEXP * LDST * 4,    wn);
    }

    for (int c = 0; c < NCHUNK; ++c) {
        const int kb = c * KC;
        if (wave == 0) {
            if (c + 1 < NCHUNK) {
                const unsigned nbuf = lds_base + ((c + 1) & 1) * (STAGE_FLTS * 4);
                tdm_load_rows(nbuf,                   wt + (kb + KC));
                tdm_load_rows(nbuf + NEXP * LDST * 4, wn + (kb + KC));
                __builtin_amdgcn_s_wait_tensorcnt(2);  // current chunk landed
            } else {
                __builtin_amdgcn_s_wait_tensorcnt(0);
            }
        }
        __syncthreads();   // buffer (c&1) visible to all waves

        if (kb + KC < HDIM) __builtin_prefetch(xrow + kb + KC, 0, 1);

        const float* wbuf = lds + (c & 1) * STAGE_FLTS;
#pragma unroll
        for (int kk = 0; kk < KC; kk += 4) {
            const float2 a2 = *(const float2*)(xrow + kb + kk);
            v2f a; a.x = a2.x; a.y = a2.y;
#pragma unroll
            for (int nt = 0; nt < 8; ++nt) {
                const float2 b2 =
                    *(const float2*)&wbuf[(nt * 16 + col) * LDST + kk + 2 * khi];
                v2f b; b.x = b2.x; b.y = b2.y;
                acc[nt] = __builtin_amdgcn_wmma_f32_16x16x4_f32(
                    false, a, false, b, (short)0, acc[nt], false, false);
            }
        }
        __syncthreads();   // everyone done reading buffer (c&1) before reuse
    }

    float nv[4];
#pragma unroll
    for (int nt = 0; nt < 4; ++nt) nv[nt] = noise[nt * 16 + col];

    // weight = logits_t + softplus(logits_n) * noise[e]; write 16x64 tile.
#pragma unroll
    for (int nt = 0; nt < 4; ++nt) {
#pragma unroll
        for (int r = 0; r < 8; ++r) {
            float z  = acc[nt + 4][r];
            float sp = fmaxf(z, 0.f) + log1pf(__expf(-fabsf(z)));
            float w  = acc[nt][r] + sp * nv[nt];
            lds[wave * TILE_WAVE + (r + 8 * khi) * TILE_ROW + nt * 16 + col] = w;
        }
    }
    if (tid < NEXP) { sh_sumw[tid] = 0.f; sh_soft[tid] = 0.f; }
    __syncthreads();

    // One thread per token: top-8 selection + masked softmax + reductions.
    if (tid < TOK_PER_BLK) {
        const float* row = &lds[(tid >> 4) * TILE_WAVE + (tid & 15) * TILE_ROW];
        unsigned long long sel = 0ull;
        float rowmax = -3.4e38f;
        for (int pass = 0; pass < 8; ++pass) {
            float best = -3.4e38f; int bi = 0;
            for (int e = 0; e < NEXP; ++e) {
                if (!((sel >> e) & 1ull)) {
                    float v = row[e];
                    if (v > best) { best = v; bi = e; }
                }
            }
            sel |= 1ull << bi;
            if (pass == 0) rowmax = best;
        }
        float Z = 0.f;
        for (int e = 0; e < NEXP; ++e)
            if ((sel >> e) & 1ull) Z += __expf(row[e] - rowmax);
        const float invZ = 1.f / Z;
        for (int e = 0; e < NEXP; ++e) {
            float w = row[e];
            atomicAdd(&sh_sumw[e], w);
            if ((sel >> e) & 1ull) atomicAdd(&sh_soft[e], __expf(w - rowmax) * invZ);
        }
    }
    __syncthreads();

    if (tid < NEXP) {
        atomicAdd(&gsum[tid],        sh_sumw[tid]);   // Sigma weight  (for mean_w)
        atomicAdd(&gsum[NEXP + tid], sh_soft[tid]);   // Sigma soft_w  (for balance loss)
    }
}

__global__ void moe_gate_finalize_91122026152203(const float* __restrict__ gsum,
                                                 float* __restrict__ out) {
    __shared__ float mw[NEXP];
    __shared__ float sw[NEXP];
    const int t = threadIdx.x;
    if (t < NEXP) {
        mw[t] = gsum[t] * (1.0f / (float)TOKENS);
        sw[t] = gsum[NEXP + t];
    }
    __syncthreads();
    if (t == 0) {
        unsigned long long sel = 0ull;
        float g[8]; int gi[8];
        for (int p = 0; p < 8; ++p) {
            float best = -3.4e38f; int bi = 0;
            for (int e = 0; e < NEXP; ++e) {
                if (!((sel >> e) & 1ull) && mw[e] > best) { best = mw[e]; bi = e; }
            }
            sel |= 1ull << bi; g[p] = best; gi[p] = bi;
        }
        const float m0 = g[0];
        float Z = 0.f;
        for (int p = 0; p < 8; ++p) Z += __expf(g[p] - m0);
        for (int p = 0; p < 8; ++p) out[p]     = __expf(g[p] - m0) / Z;
        for (int p = 0; p < 8; ++p) out[8 + p] = (float)gi[p];

        float mean = 0.f;
        for (int e = 0; e < NEXP; ++e) mean += sw[e];
        mean *= (1.0f / (float)NEXP);
        float var = 0.f;
        for (int e = 0; e < NEXP; ++e) { float d = sw[e] - mean; var += d * d; }
        var *= (1.0f / (float)(NEXP - 1));          // ddof=1 (torch-style std)
        out[16] = var / (mean * mean);              // (std/mean)^2
    }
}

extern "C" void kernel_launch(void* const* d_in, const int* in_sizes, int n_in,
                              void* d_out, int out_size, void* d_ws, size_t ws_size,
                              hipStream_t stream) {
    (void)in_sizes; (void)n_in; (void)out_size; (void)ws_size;
    const float* x     = (const float*)d_in[0];
    const float* wt    = (const float*)d_in[1];
    const float* wn    = (const float*)d_in[2];
    const float* noise = (const float*)d_in[3];
    float* ws  = (float*)d_ws;
    float* out = (float*)d_out;

    hipLaunchKernelGGL(moe_zero_ws_91122026152203, dim3(1), dim3(128), 0, stream, ws);
    hipLaunchKernelGGL(moe_gate_main_91122026152203, dim3(NBLOCKS), dim3(TPB), 0, stream,
                       x, wt, wn, noise, ws);
    hipLaunchKernelGGL(moe_gate_finalize_91122026152203, dim3(1), dim3(64), 0, stream,
                       ws, out);
}